// Model_11888469475981
// MI455X (gfx1250) — compile-verified
//
#include <hip/hip_runtime.h>

#ifndef __has_builtin
#define __has_builtin(x) 0
#endif

// Problem geometry (fixed by the reference).
#define NB 8
#define CC 32
#define DD 16        // input D; padded D = 17 (last plane all-zero)
#define HH 64
#define WW 256
#define WP 259       // 1 + 256 + 2   (W pad left=1, right=2)
#define HP 66        // 1 + 64 + 1    (H pad)
#define ROWP 264     // LDS row pitch (floats): r[i] = xp[i-3]; 264*4 = 1056 B (16B-aligned pitch)

// gfx1250 async global->LDS paths (ASYNCcnt-tracked), tier-guarded.
#if __has_builtin(__builtin_amdgcn_global_load_async_to_lds_b128)
#define ASYNC_B128 1
#else
#define ASYNC_B128 0
#endif
#if __has_builtin(__builtin_amdgcn_global_load_async_to_lds_b32)
#define ASYNC_B32 1
#else
#define ASYNC_B32 0
#endif

typedef int v4i __attribute__((vector_size(4 * sizeof(int))));
typedef __attribute__((address_space(1))) int gas_int;   // global (AS1) scalar
typedef __attribute__((address_space(3))) int las_int;   // LDS    (AS3) scalar
typedef __attribute__((address_space(1))) v4i gas_v4i;   // global (AS1) int4
typedef __attribute__((address_space(3))) v4i las_v4i;   // LDS    (AS3) int4

// LDS row layout: rv[i] = xp[i-3] with zero halo at rv[0..3] and rv[260..263].
// Returns fused[w] = xp[w] + sel(w) * softsign(xp[w]),  w in [0, 258].
__device__ __forceinline__ float fused_col(const float* __restrict__ rv, int w) {
    const float vm2 = rv[w + 1];   // xp[w-2]
    const float vm1 = rv[w + 2];   // xp[w-1]
    const float cv  = rv[w + 3];   // xp[w]
    const float vp1 = rv[w + 4];   // xp[w+1]
    const float vp2 = rv[w + 5];   // xp[w+2]
    bool sel;
    if (w & 1) {
        // w = 2l+1: middle of window l (first-occurrence argmax == 1)
        sel = (cv > vm1) & (cv >= vp1);
    } else {
        // w = 2l: pos 0 of window l  OR  pos 2 of window l-1
        sel = ((cv >= vp1) & (cv >= vp2)) | ((cv > vm2) & (cv > vm1));
    }
    // softsign via v_rcp_f32 (~1 ulp) — well within fp32 tolerance
    const float ss = cv * __builtin_amdgcn_rcpf(1.0f + __builtin_fabsf(cv));
    return cv + (sel ? ss : 0.0f);
}

__global__ __launch_bounds__(256)
void fused_pad_pool_unpool_mean(const float* __restrict__ x,
                                float* __restrict__ out) {
    const int hp = blockIdx.x;   // 0..65 (padded H)
    const int c  = blockIdx.y;   // 0..31
    const int n  = blockIdx.z;   // 0..7
    const int t  = threadIdx.x;  // 0..255

    float* orow = out + ((((size_t)n * CC + c) * HP + hp) * (size_t)WP);

    // H-pad rows: every D-plane contribution is zero -> output row is zero.
    if (hp == 0 || hp == HP - 1) {
        orow[t] = 0.0f;
        if (t < WP - WW) orow[WW + t] = 0.0f;   // cols 256..258
        return;
    }
    const int h = hp - 1;

    // LDS: 16 rows, r[d][i] = xp[i-3]; x[t] lands at r[d][t+4] (16B-aligned chunks).
    __shared__ float r[DD][ROWP];

    // Zero the constant halo: r[d][0..3] (xp[-3..0]) and r[d][260..263] (xp[257..260]).
    if (t < DD * 8) {
        const int d = t >> 3, j = t & 7;
        r[d][(j < 4) ? j : (j + 256)] = 0.0f;
    }

    // x[n][c][d][h][w]: row base for plane d; rows are 1 KB, 16B-aligned chunks.
    const size_t base0   = ((((size_t)n * CC + c) * DD) * HH + h) * (size_t)WW;
    const size_t dstride = (size_t)HH * WW;   // 16384 floats between d-planes

#if ASYNC_B128
    // 16 KB per block via 4 wide async loads per thread: chunk = t + 256k,
    // d = chunk/64, q = chunk%64 -> 512 B contiguous per wave per instruction.
#pragma unroll
    for (int k = 0; k < 4; ++k) {
        const int chunk = t + (k << 8);
        const int d = chunk >> 6;
        const int q = chunk & 63;
        __builtin_amdgcn_global_load_async_to_lds_b128(
            (gas_v4i*)(x + base0 + (size_t)d * dstride + (size_t)(q << 2)),
            (las_v4i*)&r[d][4 + (q << 2)],
            /*offset=*/0, /*cpol=*/0);
    }
#elif ASYNC_B32
#pragma unroll
    for (int d = 0; d < DD; ++d) {
        __builtin_amdgcn_global_load_async_to_lds_b32(
            (gas_int*)(x + base0 + (size_t)d * dstride + (size_t)t),
            (las_int*)&r[d][t + 4],
            /*offset=*/0, /*cpol=*/0);
    }
#endif

#if ASYNC_B128 || ASYNC_B32
#if __has_builtin(__builtin_amdgcn_s_wait_asynccnt)
    __builtin_amdgcn_s_wait_asynccnt(0);
#else
    asm volatile("s_wait_asynccnt 0" ::: "memory");
#endif
#else
    // Fallback: pipelined register loads, then LDS stores.
    float v[DD];
#pragma unroll
    for (int d = 0; d < DD; ++d) v[d] = x[base0 + (size_t)d * dstride + (size_t)t];
#pragma unroll
    for (int d = 0; d < DD; ++d) r[d][t + 4] = v[d];
#endif
    __syncthreads();

    // Accumulate fused values over the 16 real D-planes (plane 16 is zero).
    float accA = 0.0f;            // column w = t       (0..255)
    float accB = 0.0f;            // column w = 256 + t (t < 3)
#pragma unroll
    for (int d = 0; d < DD; ++d) {
        const float* rv = r[d];
        accA += fused_col(rv, t);
        if (t < WP - WW) accB += fused_col(rv, WW + t);
    }

    const float inv17 = 1.0f / 17.0f;   // mean over padded D = 17 planes
    orow[t] = accA * inv17;
    if (t < WP - WW) orow[WW + t] = accB * inv17;
}

extern "C" void kernel_launch(void* const* d_in, const int* in_sizes, int n_in,
                              void* d_out, int out_size, void* d_ws, size_t ws_size,
                              hipStream_t stream) {
    (void)in_sizes; (void)n_in; (void)out_size; (void)d_ws; (void)ws_size;
    const float* x = (const float*)d_in[0];
    float* out = (float*)d_out;

    dim3 grid(HP, CC, NB);   // one block per output row (n, c, hp)
    fused_pad_pool_unpool_mean<<<grid, 256, 0, stream>>>(x, out);
}